// Edge3DEmbedding_8409545966347
// MI455X (gfx1250) — compile-verified
//
#include <hip/hip_runtime.h>
#include <math.h>

#define BB 4
#define NN 512
#define KK 128
#define EE 768

typedef __attribute__((ext_vector_type(2))) float v2f;
typedef __attribute__((ext_vector_type(8))) float v8f;

// ---------------------------------------------------------------------------
// Kernel 1: fused geometry + gaussian edge features + j-reduction.
// One block per (b,i) row. 256 threads. Streams 256KB of edge_feature per
// block (536MB total -> HBM-write bound, ~23us floor at 23.3 TB/s).
// ---------------------------------------------------------------------------
__global__ __launch_bounds__(256) void edge_kernel(
    const float* __restrict__ pos,          // [B,N,3]
    const float* __restrict__ means,        // [1,K]
    const float* __restrict__ stds,         // [1,K]
    const float* __restrict__ mul_w,        // [NUM_EDGES,1]
    const float* __restrict__ bias_w,       // [NUM_EDGES,1]
    const int*   __restrict__ nte,          // [B,N,N,2]
    const unsigned char* __restrict__ padding_mask, // [B,N] bool
    const unsigned char* __restrict__ node_mask,    // [B,N,1] bool
    float* __restrict__ edge_feature,       // [B,N,N,K]
    float* __restrict__ delta_out,          // [B,N,N,3]
    float* __restrict__ sum_edge)           // [B*N,K] workspace
{
    const int row = blockIdx.x;             // b*N + i
    const int b   = row >> 9;               // N = 512
    const int t   = threadIdx.x;

    __shared__ float xs[256];               // x = mul*dist + bias, per j
    __shared__ float sc[256];               // 0/1 padding scale, per j
    __shared__ float facc[256];             // final k-reduction

    // per-k gaussian params (thread t owns k = t & 127 in phase 2)
    const int kk = t & 127;
    const float mean_k  = means[kk];
    const float std_k   = fabsf(stds[kk]) + 0.01f;
    const float istd_k  = 1.0f / std_k;
    const float coef_k  = rsqrtf(2.0f * 3.14159f) * istd_k;

    // pos[b,i] (uniform across block -> scalarized loads)
    const float pix = pos[row * 3 + 0];
    const float piy = pos[row * 3 + 1];
    const float piz = pos[row * 3 + 2];
    const bool  nm_i = node_mask[row] != 0;

    const int jhalf = t >> 7;               // 0 or 1: which j of each pair
    float acc = 0.0f;

    for (int tile = 0; tile < 2; ++tile) {
        const int j  = tile * 256 + t;
        const int jg = b * NN + j;

        // ---- phase 1: per-j quantities (each thread = one j) ----
        const float dx = pos[jg * 3 + 0] - pix;
        const float dy = pos[jg * 3 + 1] - piy;
        const float dz = pos[jg * 3 + 2] - piz;
        const float d2 = dx * dx + dy * dy + dz * dz;
        const float dist = (d2 > 0.0f) ? sqrtf(d2) : 0.0f;
        const float invd = 1.0f / (dist + 1e-5f);

        const size_t eidx = (size_t)row * NN + j;
        delta_out[eidx * 3 + 0] = dx * invd;
        delta_out[eidx * 3 + 1] = dy * invd;
        delta_out[eidx * 3 + 2] = dz * invd;

        const int et0 = nm_i               ? 0 : nte[eidx * 2 + 0];
        const int et1 = (node_mask[jg]!=0) ? 0 : nte[eidx * 2 + 1];
        const float mul  = mul_w[et0]  + mul_w[et1];
        const float bias = bias_w[et0] + bias_w[et1];

        __syncthreads();                    // previous tile's readers done
        xs[t] = mul * dist + bias;
        sc[t] = padding_mask[jg] ? 0.0f : 1.0f;
        __syncthreads();

        // ---- phase 2: gaussian features, coalesced stores over k ----
        float* ef = edge_feature + ((size_t)row * NN + tile * 256) * KK;
        for (int jl = jhalf; jl < 256; jl += 2) {
            const float x = xs[jl];         // LDS broadcast
            const float s = sc[jl];
            const float u = (x - mean_k) * istd_k;
            const float f = s * coef_k * __expf(-0.5f * u * u);
            acc += f;
            ef[(size_t)jl * KK + kk] = f;
        }
    }

    // reduce the two j-halves per k, write sum_edge to workspace
    __syncthreads();
    facc[t] = acc;
    __syncthreads();
    if (t < 128)
        sum_edge[(size_t)row * KK + t] = facc[t] + facc[t + 128];
}

// ---------------------------------------------------------------------------
// Kernel 2: merge = sum_edge @ proj_w^T + proj_b, masked.
// M=2048, N=768, K=128 in f32 via V_WMMA_F32_16X16X4_F32.
// Block = 8 waves; each wave owns one 16x16 tile; A tile shared via LDS.
// ---------------------------------------------------------------------------
__global__ __launch_bounds__(256) void merge_gemm_kernel(
    const float* __restrict__ sum_edge,     // [2048,128]
    const float* __restrict__ proj_w,       // [768,128]  (B^T)
    const float* __restrict__ proj_b,       // [768]
    const unsigned char* __restrict__ padding_mask, // [B*N]
    const unsigned char* __restrict__ node_mask,    // [B*N]
    float* __restrict__ merge)              // [2048,768]
{
    const int t    = threadIdx.x;
    const int wave = t >> 5;
    const int lane = t & 31;
    const int row0 = blockIdx.y * 16;
    const int col0 = blockIdx.x * 128 + wave * 16;

    __shared__ float Alds[16][130];         // padded: stride 130 dwords

    for (int idx = t; idx < 16 * 128; idx += 256) {
        const int r = idx >> 7;
        const int k = idx & 127;
        Alds[r][k] = sum_edge[(size_t)(row0 + r) * KK + k];
    }
    __syncthreads();

    const int half = lane >> 4;             // K-half: lanes 0-15 -> K={0,1}, 16-31 -> K={2,3}
    const int ln   = lane & 15;             // M (for A) / N (for B,C)

    const float* ap = &Alds[ln][2 * half];
    const float* bp = proj_w + (size_t)(col0 + ln) * KK + 2 * half;

    v8f c = {};
    #pragma unroll
    for (int k0 = 0; k0 < KK; k0 += 4) {
        v2f a, bv;
        a.x  = ap[k0];
        a.y  = ap[k0 + 1];
        bv.x = bp[k0];
        bv.y = bp[k0 + 1];
        // D = A(16x4) * B(4x16) + C   (8 args: neg_a,A,neg_b,B,c_mod,C,reuse_a,reuse_b)
        c = __builtin_amdgcn_wmma_f32_16x16x4_f32(
                false, a, false, bv, (short)0, c, false, false);
    }

    const float bias = proj_b[col0 + ln];
    #pragma unroll
    for (int v = 0; v < 8; ++v) {
        const int r = row0 + v + half * 8;  // C layout: vgpr v -> M=v (low lanes) / v+8 (high)
        const bool masked = (padding_mask[r] != 0) || (node_mask[r] != 0);
        const float val = masked ? 0.0f : (c[v] + bias);
        merge[(size_t)r * EE + (col0 + ln)] = val;
    }
}

// ---------------------------------------------------------------------------
extern "C" void kernel_launch(void* const* d_in, const int* in_sizes, int n_in,
                              void* d_out, int out_size, void* d_ws, size_t ws_size,
                              hipStream_t stream) {
    (void)in_sizes; (void)n_in; (void)out_size; (void)ws_size;

    const float* pos    = (const float*)d_in[0];
    const float* means  = (const float*)d_in[1];
    const float* stds   = (const float*)d_in[2];
    const float* mul_w  = (const float*)d_in[3];
    const float* bias_w = (const float*)d_in[4];
    const float* proj_w = (const float*)d_in[5];
    const float* proj_b = (const float*)d_in[6];
    const int*   nte    = (const int*)d_in[7];
    const unsigned char* pm = (const unsigned char*)d_in[8];
    const unsigned char* nm = (const unsigned char*)d_in[9];

    float* edge_feature = (float*)d_out;                                // [B,N,N,K]
    float* merge        = edge_feature + (size_t)BB * NN * NN * KK;     // [B,N,E]
    float* delta_out    = merge + (size_t)BB * NN * EE;                 // [B,N,N,3]
    float* sum_edge     = (float*)d_ws;                                 // [B*N,K] = 1MB

    edge_kernel<<<dim3(BB * NN), 256, 0, stream>>>(
        pos, means, stds, mul_w, bias_w, nte, pm, nm,
        edge_feature, delta_out, sum_edge);

    merge_gemm_kernel<<<dim3(EE / 128, (BB * NN) / 16), 256, 0, stream>>>(
        sum_edge, proj_w, proj_b, pm, nm, merge);
}